// Positional_Encoding_33758442946812
// MI455X (gfx1250) — compile-verified
//
#include <hip/hip_runtime.h>
#include <hip/hip_bf16.h>

// Positional-encoding table generator for MI455X (gfx1250).
// out[k, 2i]   = sin(k / 10000^(2i/d))
// out[k, 2i+1] = cos(k / 10000^(2i/d)),  d = 512.
//
// Store-bandwidth-bound: 128 MiB written, ~5.7 us floor at 23.3 TB/s.
// Strategy: one float4 (two sin/cos pairs) per lane per row, 4 rows per
// block so the per-column exp2 is amortized; hardware v_sin/v_cos with an
// f64 range reduction (angles reach 65535 rad, far outside the +/-256-rev
// valid domain of v_sin_f32).

#define PE_EMB        512
#define PE_VECS_ROW   128   // 512 floats / 4 per float4
#define PE_ROWS_BLK   4

// -2 * log2(10000) / d
#define PE_C  (-2.0f * 13.287712379549449f / 512.0f)
#define PE_INV2PI_D 0.15915494309189535  // 1/(2*pi), double

__global__ __launch_bounds__(PE_VECS_ROW)
void Positional_Encoding_33758442946812_kernel(float* __restrict__ out,
                                               int seq_len) {
    const int q  = threadIdx.x;                    // float4 column 0..127
    const int k0 = blockIdx.x * PE_ROWS_BLK;       // first row for this block

    // inv_denom = 10000^(-2i/d) = exp2(PE_C * i), i = 2q and 2q+1.
    // Computed once per thread, reused for all PE_ROWS_BLK rows.
    const float inv0 = __builtin_amdgcn_exp2f(PE_C * (float)(2 * q));
    const float inv1 = __builtin_amdgcn_exp2f(PE_C * (float)(2 * q + 1));

#pragma unroll
    for (int r = 0; r < PE_ROWS_BLK; ++r) {
        const int k = k0 + r;
        if (k >= seq_len) return;

        const float kf = (float)k;                 // k < 2^17, exact in f32
        const float a0 = kf * inv0;                // angle (radians), f32
        const float a1 = kf * inv1;

        // Range-reduce to revolutions in [0,1): r = fract(a / 2pi).
        // Done in f64 because a can be ~6.5e4 rad; f32 fract there would
        // lose ~10 bits. Cost: ~3 f64 ops per PAIR (sin & cos share it).
        const double r0 = (double)a0 * PE_INV2PI_D;
        const double r1 = (double)a1 * PE_INV2PI_D;
        const float  f0 = (float)(r0 - __builtin_floor(r0));
        const float  f1 = (float)(r1 - __builtin_floor(r1));

        float4 v;
        v.x = __builtin_amdgcn_sinf(f0);   // v_sin_f32: sin(2*pi*f0)
        v.y = __builtin_amdgcn_cosf(f0);   // v_cos_f32
        v.z = __builtin_amdgcn_sinf(f1);
        v.w = __builtin_amdgcn_cosf(f1);

        // Coalesced global_store_b128: 128 lanes x 16B = one 2 KiB row chunk.
        *(float4*)(out + (size_t)k * PE_EMB + (size_t)q * 4) = v;
    }
}

extern "C" void kernel_launch(void* const* d_in, const int* in_sizes, int n_in,
                              void* d_out, int out_size, void* d_ws, size_t ws_size,
                              hipStream_t stream) {
    (void)d_in; (void)in_sizes; (void)n_in; (void)d_ws; (void)ws_size;

    // seq_len is a device-side scalar we cannot read during graph capture;
    // recover it from the host-visible output size: out = [seq_len, 512].
    float* out = (float*)d_out;
    const int seq_len = out_size / PE_EMB;
    const int blocks  = (seq_len + PE_ROWS_BLK - 1) / PE_ROWS_BLK;

    Positional_Encoding_33758442946812_kernel<<<blocks, PE_VECS_ROW, 0, stream>>>(
        out, seq_len);
}